// SoftSpearmanLoss_53480932770568
// MI455X (gfx1250) — compile-verified
//
#include <hip/hip_runtime.h>

// ---------------------------------------------------------------------------
// SoftSpearmanLoss for MI455X (gfx1250, wave32).
// Dominant cost: B^2 * L pairwise sigmoids (compute-bound, transcendental).
// j-reduction done on the matrix pipe via V_WMMA_F32_16X16X4_F32 (A = 16x4
// tile of sigmoids, B = ones), f32 accumulate -> full precision ranks.
// ---------------------------------------------------------------------------

typedef float v8f __attribute__((ext_vector_type(8)));
typedef float v2f __attribute__((ext_vector_type(2)));

__device__ __forceinline__ float fast_sigmoid(float d) {
    // sigmoid(d) = 1 / (1 + exp(-d)) ; exp(-d) = exp2(-d * log2(e))
#if __has_builtin(__builtin_amdgcn_exp2f)
    float e = __builtin_amdgcn_exp2f(-d * 1.44269504088896340736f);
#else
    float e = __expf(-d);
#endif
#if __has_builtin(__builtin_amdgcn_rcpf)
    return __builtin_amdgcn_rcpf(1.0f + e);
#else
    return 1.0f / (1.0f + e);
#endif
}

__device__ __forceinline__ float wave_sum32(float v) {
    // butterfly reduction across wave32; all lanes end with the total
    #pragma unroll
    for (int o = 16; o > 0; o >>= 1) v += __shfl_xor(v, o, 32);
    return v;
}

// ---------------------------------------------------------------------------
// Kernel 1: soft ranks for both matrices.
// grid.x = 2 (matrix) * L (label) * chunks (B/128). 256 threads = 8 waves.
// Wave w handles rows [i0, i0+16); inner loop covers all 2048 j's, 4 per WMMA.
// ---------------------------------------------------------------------------
__global__ __launch_bounds__(256) void ssl_rank_kernel(
    const float* __restrict__ preds, const float* __restrict__ targets,
    float* __restrict__ pranks, float* __restrict__ tranks,
    int B, int L, int chunks) {
    __shared__ float s_col[2048];  // one label column (B <= 2048), 8 KB

    const int per_m  = L * chunks;
    const int m      = blockIdx.x / per_m;
    const int rem    = blockIdx.x - m * per_m;
    const int l      = rem / chunks;
    const int chunk  = rem - l * chunks;

    const float* __restrict__ src = (m == 0) ? preds  : targets;
    float*       __restrict__ dst = (m == 0) ? pranks : tranks;

#if __has_builtin(__builtin_prefetch)
    __builtin_prefetch(&src[l], 0, 1);   // gfx1250: global_prefetch_b8
#endif

    // Stage column into LDS; sigmoid(preds) is applied here once (m==0).
    for (int j = threadIdx.x; j < B; j += blockDim.x) {
        float v = src[(size_t)j * L + l];
        if (m == 0) v = fast_sigmoid(v);
        s_col[j] = v;
    }
    __syncthreads();

    const int lane = threadIdx.x & 31;
    const int wv   = threadIdx.x >> 5;
    const int i0   = chunk * 128 + wv * 16;

    // A-matrix (16x4 f32) layout: lanes 0-15 -> rows, K={0,1};
    //                             lanes 16-31 -> rows, K={2,3}.
    const float xi   = s_col[i0 + (lane & 15)];
    const int   joff = (lane & 16) >> 3;  // 0 for lanes 0-15, 2 for 16-31

    v2f bones; bones.x = 1.0f; bones.y = 1.0f;   // B = ones (4x16)
    v8f c = {0.0f, 0.0f, 0.0f, 0.0f, 0.0f, 0.0f, 0.0f, 0.0f};

    for (int j = 0; j < B; j += 4) {
        v2f a;
        a.x = fast_sigmoid(xi - s_col[j + joff]);      // LDS broadcast reads
        a.y = fast_sigmoid(xi - s_col[j + joff + 1]);
        // D[m][n] = sum_k A[m][k] * 1 + C  -> every column = row-sum chunk
        c = __builtin_amdgcn_wmma_f32_16x16x4_f32(
                /*neg_a=*/false, a, /*neg_b=*/false, bones,
                /*c_mod=*/(short)0, c, /*reuse_a=*/false, /*reuse_b=*/false);
    }

    // D layout: VGPR r, lane 0  -> D[r][0]   (rows i0..i0+7)
    //           VGPR r, lane 16 -> D[8+r][0] (rows i0+8..i0+15)
    #pragma unroll
    for (int r = 0; r < 8; ++r) {
        if (lane == 0)  dst[(size_t)l * B + i0 + r]     = c[r];
        if (lane == 16) dst[(size_t)l * B + i0 + 8 + r] = c[r];
    }
}

// ---------------------------------------------------------------------------
// Kernel 2: per-label Pearson over ranks (two-pass, centered), target-std
// mask (ddof=1), masked mean -> loss. One wave32 per label; 1 block.
// ---------------------------------------------------------------------------
__global__ __launch_bounds__(1024) void ssl_finalize_kernel(
    const float* __restrict__ targets,
    const float* __restrict__ pranks, const float* __restrict__ tranks,
    float* __restrict__ out, int B, int L) {
    __shared__ float s_corr[32];
    __shared__ float s_mask[32];

    const int lane = threadIdx.x & 31;
    const int w    = threadIdx.x >> 5;

    float corr_m = 0.0f, msk = 0.0f;
    if (w < L) {
        const int l = w;
        // pass 1: means of ranks + raw sums for target std
        float s_pr = 0.0f, s_tr = 0.0f, s_t = 0.0f, s_t2 = 0.0f;
        for (int i = lane; i < B; i += 32) {
            float pr = pranks[(size_t)l * B + i];
            float tr = tranks[(size_t)l * B + i];
            float t  = targets[(size_t)i * L + l];
            s_pr += pr; s_tr += tr; s_t += t; s_t2 += t * t;
        }
        s_pr = wave_sum32(s_pr); s_tr = wave_sum32(s_tr);
        s_t  = wave_sum32(s_t);  s_t2 = wave_sum32(s_t2);

        const float invB = 1.0f / (float)B;
        const float mpr = s_pr * invB, mtr = s_tr * invB;

        // pass 2: centered second moments (avoids catastrophic cancellation)
        float sxy = 0.0f, sxx = 0.0f, syy = 0.0f;
        for (int i = lane; i < B; i += 32) {
            float pc = pranks[(size_t)l * B + i] - mpr;
            float tc = tranks[(size_t)l * B + i] - mtr;
            sxy += pc * tc; sxx += pc * pc; syy += tc * tc;
        }
        sxy = wave_sum32(sxy); sxx = wave_sum32(sxx); syy = wave_sum32(syy);

        float corr = sxy / sqrtf(sxx * syy + 1e-8f);
        corr = fminf(1.0f, fmaxf(-1.0f, corr));

        float var = (s_t2 - s_t * s_t * invB) / (float)(B - 1);  // ddof=1
        float sd  = sqrtf(fmaxf(var, 0.0f));
        msk    = (sd >= 1e-8f) ? 1.0f : 0.0f;
        corr_m = corr * msk;
    }
    if (lane == 0) { s_corr[w] = corr_m; s_mask[w] = msk; }
    __syncthreads();

    if (threadIdx.x == 0) {
        float sc = 0.0f, cnt = 0.0f;
        for (int k = 0; k < 32; ++k) { sc += s_corr[k]; cnt += s_mask[k]; }
        float mean = sc / fmaxf(cnt, 1.0f);
        out[0] = (cnt > 0.0f) ? (1.0f - mean) : 0.0f;
    }
}

// ---------------------------------------------------------------------------
extern "C" void kernel_launch(void* const* d_in, const int* in_sizes, int n_in,
                              void* d_out, int out_size, void* d_ws, size_t ws_size,
                              hipStream_t stream) {
    const int L = 30;
    const int B = in_sizes[0] / L;   // 2048

    const float* preds   = (const float*)d_in[0];
    const float* targets = (const float*)d_in[1];

    float* pranks = (float*)d_ws;                    // [L][B] ranks of sigmoid(preds)
    float* tranks = pranks + (size_t)B * L;          // [L][B] ranks of targets

    const int chunks = B / 128;                      // 128 rows per block (8 waves x 16)
    dim3 grid(2 * L * chunks);                       // 960 blocks
    ssl_rank_kernel<<<grid, 256, 0, stream>>>(preds, targets, pranks, tranks,
                                              B, L, chunks);

    ssl_finalize_kernel<<<1, 1024, 0, stream>>>(targets, pranks, tranks,
                                                (float*)d_out, B, L);
}